// MetriplecticPropagator_55155970015889
// MI455X (gfx1250) — compile-verified
//
#include <hip/hip_runtime.h>
#include <math.h>

// ---------------------------------------------------------------------------
// MetriplecticPropagator forward for MI455X (gfx1250, wave32).
//   u_flat  (4*128*65, 8, 64) complex64   -> d_out[0 .. 34,078,720) floats
//   A_flat  same shape                    -> d_out[34,078,720 .. 68,157,440)
//   dt_eff  (2,8,64,1,1) float32          -> d_out[68,157,440 .. +1024)
// Strategy: rfft2 as fp32 DFT GEMMs on V_WMMA_F32_16X16X4_F32; TDM tensor
// store for the transposed output scatter; everything else bandwidth kernels.
// ---------------------------------------------------------------------------

typedef float v2f  __attribute__((ext_vector_type(2)));
typedef float v8f  __attribute__((ext_vector_type(8)));
typedef unsigned int u32x4 __attribute__((ext_vector_type(4)));
typedef int   i32x4 __attribute__((ext_vector_type(4)));
typedef int   i32x8 __attribute__((ext_vector_type(8)));

// ---- workspace layout (floats). total ~235 KB, assumed <= ws_size ---------
static constexpr int OFF_CWT   = 0;      // 80x128 col-major cos  (W-DFT)
static constexpr int OFF_SWT   = 10240;  // 80x128 col-major -sin (W-DFT)
static constexpr int OFF_CH    = 20480;  // 128x128 row-major cos/128  (H-DFT)
static constexpr int OFF_SH    = 36864;  // 128x128 row-major -sin/128 (H-DFT)
static constexpr int OFF_GE    = 53248;  // 1024: mean |z| per (b,t,c)
static constexpr int OFF_POOL  = 54272;  // 256: conv pooled sums
static constexpr int OFF_DTE   = 54528;  // 1024 dt_eff
static constexpr int OFF_RR    = 55552;  // 1024 rot real
static constexpr int OFF_RI    = 56576;  // 1024 rot imag
static constexpr int OFF_BV    = 57600;  // 1024 B_valve
static constexpr int OFF_GB    = 58624;  // 64 gamma_base
static constexpr int OFF_NU    = 58688;  // 1 nu
static constexpr int WS_FLOATS = 58689;

static constexpr unsigned long long U_FLOATS = 34078720ull;   // 4*8320*8*64*2
static constexpr unsigned long long A_OFF    = U_FLOATS;
static constexpr unsigned long long DT_OFF   = 2ull * U_FLOATS;

__device__ __forceinline__ float sigm(float x) { return 1.f / (1.f + __expf(-x)); }

// ---------------------------------------------------------------------------
// Kernel 0: generate DFT matrices (ortho norm folded into H stage), zero misc.
// ---------------------------------------------------------------------------
__global__ __launch_bounds__(256) void k_setup(float* ws) {
  const float PI_64 = 0.0490873852123405f;  // 2*pi/128
  for (int idx = blockIdx.x * 256 + threadIdx.x; idx < WS_FLOATS;
       idx += gridDim.x * 256) {
    if (idx < 10240) {                       // CwT[w][j] = cos(2pi j w/128)
      int w = idx >> 7, j = idx & 127; int p = (j * w) & 127;
      ws[idx] = cosf((float)p * PI_64);
    } else if (idx < 20480) {                // SwT[w][j] = -sin(...)
      int q = idx - 10240; int w = q >> 7, j = q & 127; int p = (j * w) & 127;
      ws[idx] = -sinf((float)p * PI_64);
    } else if (idx < 36864) {                // Ch[m][h] = cos(...)/128
      int q = idx - 20480; int m = q >> 7, h = q & 127; int p = (m * h) & 127;
      ws[idx] = cosf((float)p * PI_64) * (1.f / 128.f);
    } else if (idx < 53248) {                // Sh[m][h] = -sin(...)/128
      int q = idx - 36864; int m = q >> 7, h = q & 127; int p = (m * h) & 127;
      ws[idx] = -sinf((float)p * PI_64) * (1.f / 128.f);
    } else {
      ws[idx] = 0.f;                         // pooled accumulators etc.
    }
  }
}

// ---------------------------------------------------------------------------
// Kernel 1: ge[b,t,c] = mean over HxW of |z| (== v_proxy - 1e-6)
// ---------------------------------------------------------------------------
__global__ __launch_bounds__(256) void k_reduce(const float* __restrict__ zr,
                                                const float* __restrict__ zi,
                                                float* __restrict__ ge) {
  __shared__ float red[256];
  const int n = blockIdx.x;  // 1024 = (b*T+t)*C+c
  const float* pr = zr + (size_t)n * 16384;
  const float* pi = zi + (size_t)n * 16384;
  float s = 0.f;
  for (int i = threadIdx.x; i < 16384; i += 256) {
    float a = pr[i], b = pi[i];
    s += sqrtf(fmaf(a, a, b * b));
  }
  red[threadIdx.x] = s;
  __syncthreads();
  for (int off = 128; off > 0; off >>= 1) {
    if (threadIdx.x < off) red[threadIdx.x] += red[threadIdx.x + off];
    __syncthreads();
  }
  if (threadIdx.x == 0) ge[n] = red[0] * (1.f / 16384.f);
}

// ---------------------------------------------------------------------------
// Kernel 2: 3x3 conv (128->16 ch) + SiLU + spatial-sum into pooled[bt][oc].
// Block = (bt, h); thread = w. Weight loads are wave-uniform (scalar path).
// ---------------------------------------------------------------------------
__global__ __launch_bounds__(128) void k_conv(const float* __restrict__ zr,
                                              const float* __restrict__ zi,
                                              const float* __restrict__ cw,
                                              const float* __restrict__ cb,
                                              float* __restrict__ pooled) {
  const int bt = blockIdx.x >> 7;
  const int h  = blockIdx.x & 127;
  const int w  = threadIdx.x;
  float acc[16];
#pragma unroll
  for (int o = 0; o < 16; ++o) acc[o] = 0.f;
  for (int ic = 0; ic < 128; ++ic) {
    const float* sp = (ic < 64 ? zr : zi) + ((size_t)bt * 64 + (ic & 63)) * 16384;
    float v[9];
#pragma unroll
    for (int ky = 0; ky < 3; ++ky) {
      int hh = h + ky - 1;
#pragma unroll
      for (int kx = 0; kx < 3; ++kx) {
        int ww = w + kx - 1;
        float val = 0.f;
        if (hh >= 0 && hh < 128 && ww >= 0 && ww < 128) val = sp[hh * 128 + ww];
        v[ky * 3 + kx] = val;
      }
    }
    const float* wp = cw + (size_t)ic * 9;  // conv_w[oc][ic][3][3]
#pragma unroll
    for (int o = 0; o < 16; ++o) {
      const float* wo = wp + (size_t)o * 1152;
      float s = 0.f;
#pragma unroll
      for (int q = 0; q < 9; ++q) s = fmaf(v[q], wo[q], s);
      acc[o] += s;
    }
  }
#pragma unroll
  for (int o = 0; o < 16; ++o) {
    float x = acc[o] + cb[o];
    float s = x * sigm(x);                       // SiLU
    for (int off = 16; off > 0; off >>= 1)       // wave32 reduce
      s += __shfl_down(s, off, 32);
    if ((threadIdx.x & 31) == 0) atomicAdd(&pooled[bt * 16 + o], s);
  }
}

// ---------------------------------------------------------------------------
// Kernel 3: tiny — alpha, gate MLP, dt_eff, rotation scalars, B_valve, gamma.
// ---------------------------------------------------------------------------
__global__ __launch_bounds__(256) void k_small(
    float* ws, const float* __restrict__ dtp, const float* __restrict__ lin_w,
    const float* __restrict__ lin_b, const float* __restrict__ w1,
    const float* __restrict__ b1, const float* __restrict__ w2,
    const float* __restrict__ b2, const float* __restrict__ Lg,
    const float* __restrict__ diss, const float* __restrict__ visc,
    float* __restrict__ out_dt) {
  __shared__ float hid[16 * 32];
  __shared__ float sAlpha[16];
  const int tid = threadIdx.x;
  const float* ge = ws + OFF_GE;
  const float* pooled = ws + OFF_POOL;
  if (tid < 16) {
    float s = lin_b[0];
    for (int o = 0; o < 16; ++o)
      s = fmaf(pooled[tid * 16 + o] * (1.f / 16384.f), lin_w[o], s);
    sAlpha[tid] = sigm(s);
  }
  if (tid < 64) ws[OFF_GB + tid] = -logf(1.f + __expf(diss[tid]));  // -softplus
  if (tid == 64) ws[OFF_NU] = logf(1.f + __expf(visc[0]));          // softplus
  for (int i = tid; i < 512; i += 256) {        // hidden = silu(ge @ w1^T + b1)
    int bt = i >> 5, j = i & 31;
    float s = b1[j];
    for (int cc = 0; cc < 64; ++cc) s = fmaf(ge[bt * 64 + cc], w1[j * 64 + cc], s);
    hid[i] = s * sigm(s);
  }
  __syncthreads();
  const float dtv = dtp[0];
  for (int i = tid; i < 1024; i += 256) {
    int bt = i >> 6, cc = i & 63;
    float s0 = b2[cc], s2 = b2[128 + cc];
    for (int j = 0; j < 32; ++j) {
      float hv = hid[bt * 32 + j];
      s0 = fmaf(hv, w2[cc * 32 + j], s0);
      s2 = fmaf(hv, w2[(128 + cc) * 32 + j], s2);
    }
    float omega = Lg[cc] + tanhf(s0) * 0.1f;
    float bvv = 2.f * sigm(tanhf(s2));
    float g = ge[i];
    float dtmax = 1.f / ((g + 1e-6f) * 10.f);
    float dte = sAlpha[bt] * fminf(dtv, dtmax);
    float hf = 0.5f * omega * dte;               // A_rot = (1+ih)/(1-ih)
    float den = 1.f + hf * hf;
    ws[OFF_DTE + i] = dte;
    ws[OFF_RR + i] = (1.f - hf * hf) / den;
    ws[OFF_RI + i] = 2.f * hf / den;
    ws[OFF_BV + i] = bvv;
    out_dt[i] = dte;
  }
}

// ---------------------------------------------------------------------------
// Kernel 4: A_flat (pure bandwidth, 136 MB coalesced writes).
// ---------------------------------------------------------------------------
__global__ __launch_bounds__(256) void k_aflat(const float* __restrict__ ws,
                                               float* __restrict__ out) {
  const int blk = blockIdx.x;                 // 33280 rows = b4*8320 + h*65 + w
  const int b4 = blk / 8320;
  const int rem = blk - b4 * 8320;
  const int h = rem / 65;
  const int w = rem - h * 65;
  const float kyv = (float)(h < 64 ? h : h - 128) * (1.f / 128.f);
  const float kxv = (float)w * (1.f / 128.f);
  const float ksq = kxv * kxv + kyv * kyv;
  const int bbase = (b4 & 1) * 512;
  const float nu = ws[OFF_NU];
  float* outp = out + A_OFF + (unsigned long long)blk * 1024ull;
  __builtin_prefetch(&ws[OFF_DTE + bbase], 0, 0);
  for (int i = threadIdx.x; i < 512; i += 256) {  // i = t*64 + c
    int cc = i & 63;
    int n = bbase + i;
    float dte = ws[OFF_DTE + n];
    float A = __expf((ws[OFF_GB + cc] - nu * ksq) * dte);
    outp[2 * i]     = A * ws[OFF_RR + n];
    outp[2 * i + 1] = A * ws[OFF_RI + n];
  }
}

// ---------------------------------------------------------------------------
// Kernel 5: rfft2 * B_valve via WMMA-F32 DFT GEMMs + TDM tensor store.
// One block per image (b4,t,c); 8 waves; 5 column tiles of 16.
//   Stage 1: Y(128x16) = Z(128x128) @ Fw(:,tile)     (2 real GEMMs)
//   Stage 2: U(128x16) = Fh(128x128) @ Y             (4 real GEMMs)
// Y is staged TRANSPOSED in LDS ([col][row], stride 132) so stage-2
// B-fragments are contiguous K-pairs -> single ds_load_b64 per fragment reg,
// bank-conflict-free (132 mod 64 = 4).
// ---------------------------------------------------------------------------
#if __has_builtin(__builtin_amdgcn_tensor_store_from_lds) && \
    __has_builtin(__builtin_amdgcn_s_wait_tensorcnt)
#define USE_TDM_STORE 1
#if defined(__AMDGCN__)
#warning CDNA5_PROBE: TDM tensor_store_from_lds path ENABLED (device pass)
#if __clang_major__ >= 23
#warning CDNA5_PROBE: using 6-arg tensor builtin (clang >= 23)
#else
#warning CDNA5_PROBE: using 5-arg tensor builtin (clang <= 22)
#endif
#endif
#else
#if defined(__AMDGCN__)
#warning CDNA5_PROBE: TDM builtin NOT found - manual scatter fallback (device pass)
#endif
#endif

static constexpr int YSTR = 132;  // padded row length for transposed Y

__global__ __launch_bounds__(256) void k_fft(const float* __restrict__ zr,
                                             const float* __restrict__ zi,
                                             const float* __restrict__ ws,
                                             float* __restrict__ out) {
  __shared__ float sYrT[16 * YSTR];   // [col][row]
  __shared__ float sYiT[16 * YSTR];
  __shared__ float2 sU[128 * 16];

  const float* cwT = ws + OFF_CWT;
  const float* swT = ws + OFF_SWT;
  const float* chm = ws + OFF_CH;
  const float* shm = ws + OFF_SH;

  const int blk = blockIdx.x;           // 2048 = 4 * 8 * 64
  const int b4 = blk >> 9;
  const int rem = blk & 511;
  const int t = rem >> 6;
  const int c = rem & 63;
  const int nidx = ((b4 & 1) * 8 + t) * 64 + c;
  const float* src = (b4 < 2 ? zr : zi) + (size_t)nidx * 16384;
  const float bv = ws[OFF_BV + nidx];

  const int lane = threadIdx.x & 31;
  const int wv = threadIdx.x >> 5;      // 8 waves == 8 row tiles
  const int mr = lane & 15;
  const int hlf = lane >> 4;
  const int koff = hlf * 2;
  const int m0 = wv * 16;

  __builtin_prefetch(src + (size_t)threadIdx.x * 64, 0, 0);

  for (int nt = 0; nt < 5; ++nt) {
    const int n0 = nt * 16;
    // ---- Stage 1: Y = Z @ Fw(:, n0:n0+16) --------------------------------
    v8f aR = {}; v8f aI = {};
#pragma unroll 4
    for (int kk = 0; kk < 128; kk += 4) {
      const int k = kk + koff;
      v2f af; af.x = src[(m0 + mr) * 128 + k]; af.y = src[(m0 + mr) * 128 + k + 1];
      v2f bc; bc.x = cwT[(n0 + mr) * 128 + k]; bc.y = cwT[(n0 + mr) * 128 + k + 1];
      v2f bs; bs.x = swT[(n0 + mr) * 128 + k]; bs.y = swT[(n0 + mr) * 128 + k + 1];
      aR = __builtin_amdgcn_wmma_f32_16x16x4_f32(false, af, false, bc, (short)0, aR, false, false);
      aI = __builtin_amdgcn_wmma_f32_16x16x4_f32(false, af, false, bs, (short)0, aI, false, false);
    }
#pragma unroll
    for (int r = 0; r < 8; ++r) {       // transposed store: [col][row]
      sYrT[mr * YSTR + (m0 + r + hlf * 8)] = aR[r];
      sYiT[mr * YSTR + (m0 + r + hlf * 8)] = aI[r];
    }
    __syncthreads();

    // ---- Stage 2: U = Fh @ Y  (complex) ----------------------------------
    v8f uR = {}; v8f uI = {};
#pragma unroll 4
    for (int kk = 0; kk < 128; kk += 4) {
      const int k = kk + koff;
      v2f ac; ac.x = chm[(m0 + mr) * 128 + k]; ac.y = chm[(m0 + mr) * 128 + k + 1];
      v2f as; as.x = shm[(m0 + mr) * 128 + k]; as.y = shm[(m0 + mr) * 128 + k + 1];
      v2f an; an.x = -as.x; an.y = -as.y;
      v2f br; br.x = sYrT[mr * YSTR + k]; br.y = sYrT[mr * YSTR + k + 1];
      v2f bi; bi.x = sYiT[mr * YSTR + k]; bi.y = sYiT[mr * YSTR + k + 1];
      uR = __builtin_amdgcn_wmma_f32_16x16x4_f32(false, ac, false, br, (short)0, uR, false, false);
      uR = __builtin_amdgcn_wmma_f32_16x16x4_f32(false, an, false, bi, (short)0, uR, false, false);
      uI = __builtin_amdgcn_wmma_f32_16x16x4_f32(false, ac, false, bi, (short)0, uI, false, false);
      uI = __builtin_amdgcn_wmma_f32_16x16x4_f32(false, as, false, br, (short)0, uI, false, false);
    }
#pragma unroll
    for (int r = 0; r < 8; ++r) {
      const int row = m0 + r + hlf * 8;        // frequency index h
      sU[row * 16 + mr] = make_float2(uR[r] * bv, uI[r] * bv);
    }
    __syncthreads();

    // ---- write U tile to u_flat[(b4*8320 + h*65 + w)][t][c] --------------
#ifdef USE_TDM_STORE
    if (threadIdx.x < 32) {
      // 3-D tensor store: dim0 = 1 elem (8B), dim1 = w (16, stride T*C),
      // dim2 = h (128, stride 65*T*C). tensor_dim1 = 65-n0 trims w >= 65.
      unsigned long long gaddr =
          (unsigned long long)(uintptr_t)out +
          ((((unsigned long long)b4 * 8320ull + (unsigned long long)n0) * 8ull +
            (unsigned long long)t) * 64ull + (unsigned long long)c) * 8ull;
      unsigned lds_off = (unsigned)(uintptr_t)&sU[0];
      u32x4 g0;
      g0[0] = 1u;                                   // count = 1
      g0[1] = lds_off;                              // lds_addr
      g0[2] = (unsigned)gaddr;                      // global_addr[31:0]
      g0[3] = (unsigned)((gaddr >> 32) & 0x1FFFFFFull) | (2u << 30);  // type=2
      i32x8 g1;
      g1[0] = (int)(3u << 16);                      // data_size = 8 bytes
      g1[1] = (int)(1u << 16);                      // tensor_dim0 = 1
      g1[2] = (int)((unsigned)(65 - n0) << 16);     // tensor_dim1 (w bound)
      g1[3] = (int)(1u << 16);                      // tile_dim0 = 1
      g1[4] = (int)(16u | (128u << 16));            // tile_dim1=16, tile_dim2=128
      g1[5] = 512;                                  // tensor_dim0_stride = T*C
      g1[6] = (int)(33280u << 16);                  // tensor_dim1_stride = 65*512
      g1[7] = 0;
      i32x4 g2; g2[0] = 128; g2[1] = 0; g2[2] = 0; g2[3] = 0;  // tensor_dim2
      i32x4 g3; g3[0] = 0; g3[1] = 0; g3[2] = 0; g3[3] = 0;
#if __clang_major__ >= 23
      i32x8 gz; gz[0]=0; gz[1]=0; gz[2]=0; gz[3]=0; gz[4]=0; gz[5]=0; gz[6]=0; gz[7]=0;
      __builtin_amdgcn_tensor_store_from_lds(g0, g1, g2, g3, gz, 0);
#else
      __builtin_amdgcn_tensor_store_from_lds(g0, g1, g2, g3, 0);
#endif
      __builtin_amdgcn_s_wait_tensorcnt(0);
    }
#else
    for (int i = threadIdx.x; i < 128 * 16; i += 256) {
      const int hh = i >> 4;
      const int wcol = n0 + (i & 15);
      if (wcol < 65) {
        unsigned long long e =
            ((((unsigned long long)b4 * 8320ull + (unsigned long long)hh * 65ull +
               (unsigned long long)wcol) * 8ull + (unsigned long long)t) * 64ull +
             (unsigned long long)c);
        ((float2*)out)[e] = sU[i];
      }
    }
#endif
    __syncthreads();
  }
}

// ---------------------------------------------------------------------------
extern "C" void kernel_launch(void* const* d_in, const int* in_sizes, int n_in,
                              void* d_out, int out_size, void* d_ws, size_t ws_size,
                              hipStream_t stream) {
  const float* zr   = (const float*)d_in[0];
  const float* zi   = (const float*)d_in[1];
  const float* dtp  = (const float*)d_in[2];
  const float* cw   = (const float*)d_in[3];
  const float* cb   = (const float*)d_in[4];
  const float* linw = (const float*)d_in[5];
  const float* linb = (const float*)d_in[6];
  const float* w1   = (const float*)d_in[7];
  const float* b1   = (const float*)d_in[8];
  const float* w2   = (const float*)d_in[9];
  const float* b2   = (const float*)d_in[10];
  const float* Lg   = (const float*)d_in[11];
  const float* diss = (const float*)d_in[12];
  const float* visc = (const float*)d_in[13];
  float* out = (float*)d_out;
  float* ws  = (float*)d_ws;  // requires ~235 KB

  k_setup<<<dim3(232), dim3(256), 0, stream>>>(ws);
  k_reduce<<<dim3(1024), dim3(256), 0, stream>>>(zr, zi, ws + OFF_GE);
  k_conv<<<dim3(2048), dim3(128), 0, stream>>>(zr, zi, cw, cb, ws + OFF_POOL);
  k_small<<<dim3(1), dim3(256), 0, stream>>>(ws, dtp, linw, linb, w1, b1, w2, b2,
                                             Lg, diss, visc, out + DT_OFF);
  k_aflat<<<dim3(33280), dim3(256), 0, stream>>>(ws, out);
  k_fft<<<dim3(2048), dim3(256), 0, stream>>>(zr, zi, ws, out);
}